// EstimationSTD_63909113364757
// MI455X (gfx1250) — compile-verified
//
#include <hip/hip_runtime.h>
#include <math.h>

// Problem geometry (fixed by the reference): x is (16, 1, 3, 1024, 1024) f32.
#define W    1024
#define H    1024
#define BS   16
#define HW   (H * W)              // frame stride (elements)
#define BSTR (3 * HW)             // batch stride (elements), c == 1
#define ROWS_TOTAL      (BS * H)  // 16384 rows of sout
#define ROWS_PER_BLOCK  32
#define NPART           (ROWS_TOTAL / ROWS_PER_BLOCK)  // 512 partial sets

typedef float v4f __attribute__((ext_vector_type(4)));

__device__ __forceinline__ v4f v4min(v4f a, v4f b) {
    v4f r; r.x = fminf(a.x, b.x); r.y = fminf(a.y, b.y);
    r.z = fminf(a.z, b.z); r.w = fminf(a.w, b.w); return r;
}
__device__ __forceinline__ v4f v4max(v4f a, v4f b) {
    v4f r; r.x = fmaxf(a.x, b.x); r.y = fmaxf(a.y, b.y);
    r.z = fmaxf(a.z, b.z); r.w = fmaxf(a.w, b.w); return r;
}

// Pass 1: per-block column min/max partials over ROWS_PER_BLOCK rows.
// Block k covers rows [k*32, k*32+32); thread t covers columns [4t, 4t+4).
// No atomics (same-address RMW serialization at L2 would cost more than the
// whole kernel); partials go to the workspace and are folded in pass 2.
// Straight-line hot loop: no divergent guards, unroll 8 for load MLP.
__global__ __launch_bounds__(256) void estd_partial(const float* __restrict__ x,
                                                    float* __restrict__ pmn,
                                                    float* __restrict__ pmx) {
    const int t    = threadIdx.x;          // 0..255
    const int blk  = blockIdx.x;           // 0..NPART-1
    const int col  = t << 2;               // 0..1020 step 4
    const int r0   = blk * ROWS_PER_BLOCK;
    const int b    = r0 >> 10;             // batch; uniform per block (32 | 1024)
    const int rr0  = r0 & (H - 1);
    const float* base = x + (size_t)b * BSTR + (size_t)rr0 * W + col;

    v4f mn; mn.x = mn.y = mn.z = mn.w =  INFINITY;
    v4f mx; mx.x = mx.y = mx.z = mx.w = -INFINITY;

    if (b == 0) {
        // Rows of batch 0: sout = frame2 - frame0 (second difference collapses).
        #pragma unroll 8
        for (int i = 0; i < ROWS_PER_BLOCK; ++i) {
            const float* p = base + (size_t)i * W;
            v4f a = *(const v4f*)p;             // frame 0
            v4f c = *(const v4f*)(p + 2 * HW);  // frame 2
            v4f v; v.x = c.x - a.x; v.y = c.y - a.y;
                   v.z = c.z - a.z; v.w = c.w - a.w;
            mn = v4min(mn, v); mx = v4max(mx, v);
        }
    } else {
        // Rows >= h keep the frame-0 values (faithful to the reference).
        #pragma unroll 8
        for (int i = 0; i < ROWS_PER_BLOCK; ++i) {
            v4f v = *(const v4f*)(base + (size_t)i * W);
            mn = v4min(mn, v); mx = v4max(mx, v);
        }
    }

    *(v4f*)(pmn + (size_t)blk * W + col) = mn;
    *(v4f*)(pmx + (size_t)blk * W + col) = mx;
}

// Pass 2: fold NPART partials per column; emit mn and 1/denom (sklearn
// _handle_zeros_in_scale: denom = 1 where range == 0). ~4 MiB of reads, trivial.
__global__ __launch_bounds__(256) void estd_final(const float* __restrict__ pmn,
                                                  const float* __restrict__ pmx,
                                                  float* __restrict__ mn_out,
                                                  float* __restrict__ rden_out) {
    const int col = blockIdx.x * 256 + threadIdx.x;  // grid = W/256 = 4
    float mn =  INFINITY;
    float mx = -INFINITY;
    #pragma unroll 8
    for (int p = 0; p < NPART; ++p) {                // coalesced: unit stride across threads, stride W across p
        mn = fminf(mn, pmn[(size_t)p * W + col]);
        mx = fmaxf(mx, pmx[(size_t)p * W + col]);
    }
    const float rng = mx - mn;
    const float den = (rng == 0.0f) ? 1.0f : rng;
    mn_out[col]   = mn;
    rden_out[col] = 1.0f / den;
}

// Pass 3: recompute sout (re-reads hit L2: 68 MiB working set << 192 MB L2),
// normalize, and store the 64 MiB output with a non-temporal hint so it does
// not evict the inputs from L2 mid-pass.
__global__ __launch_bounds__(256) void estd_norm(const float* __restrict__ x,
                                                 const float* __restrict__ mn_arr,
                                                 const float* __restrict__ rden_arr,
                                                 float* __restrict__ out) {
    const int t   = threadIdx.x;
    const int row = blockIdx.x;             // 0..16383
    const int col = t << 2;
    const int b   = row >> 10;
    const int rr  = row & (H - 1);
    const float* p = x + (size_t)b * BSTR + (size_t)rr * W + col;

    v4f v = *(const v4f*)p;                 // frame 0
    if (b == 0) {                           // block-uniform branch
        v4f c = *(const v4f*)(p + 2 * HW);  // frame 2
        v.x = c.x - v.x; v.y = c.y - v.y; v.z = c.z - v.z; v.w = c.w - v.w;
    }
    const v4f mn = *(const v4f*)(mn_arr + col);
    const v4f rd = *(const v4f*)(rden_arr + col);
    v4f o;
    o.x = (v.x - mn.x) * rd.x;
    o.y = (v.y - mn.y) * rd.y;
    o.z = (v.z - mn.z) * rd.z;
    o.w = (v.w - mn.w) * rd.w;
    __builtin_nontemporal_store(o, (v4f*)(out + (size_t)row * W + col));
}

extern "C" void kernel_launch(void* const* d_in, const int* in_sizes, int n_in,
                              void* d_out, int out_size, void* d_ws, size_t ws_size,
                              hipStream_t stream) {
    (void)in_sizes; (void)n_in; (void)out_size; (void)ws_size;
    const float* x   = (const float*)d_in[0];   // (16,1,3,1024,1024) f32
    float*       out = (float*)d_out;           // (16,1,1024,1024)  f32

    // Workspace layout (fully overwritten each call before being read):
    //   pmn : NPART*W floats (2 MiB)
    //   pmx : NPART*W floats (2 MiB)
    //   mn  : W floats (4 KiB)
    //   rden: W floats (4 KiB)
    float* pmn  = (float*)d_ws;
    float* pmx  = pmn + (size_t)NPART * W;
    float* mn   = pmx + (size_t)NPART * W;
    float* rden = mn + W;

    estd_partial<<<NPART, 256, 0, stream>>>(x, pmn, pmx);
    estd_final<<<W / 256, 256, 0, stream>>>(pmn, pmx, mn, rden);
    estd_norm<<<ROWS_TOTAL, 256, 0, stream>>>(x, mn, rden, out);
}